// OhemCELoss_10522669875607
// MI455X (gfx1250) — compile-verified
//
#include <hip/hip_runtime.h>
#include <hip/hip_bf16.h>

// OHEM cross-entropy for MI455X (gfx1250, wave32).
// Strategy: replace the global sort with an exact 4-pass radix-select on the
// nonnegative CE float bit patterns. Per-wave reductions use the matrix pipe
// (V_WMMA_F32_16X16X4_F32 with B=ones) instead of a shuffle tree.

#define THRESH_F 0.35667494393873245f   // -log(0.7)
#define NCLASS   19
#define IGNORE_L 255

typedef __attribute__((ext_vector_type(2))) float v2f;
typedef __attribute__((ext_vector_type(8))) float v8f;

// Wave32 sum via WMMA: A = (x, 0) per lane -> row m of A is {x[m],0,x[m+16],0};
// with B = ones, D[m][n] = x[m] + x[m+16]. Summing C vgprs 0..7 gives
// lanes 0-15: sum rows 0..7, lanes 16-31: sum rows 8..15; xor-16 combine.
__device__ __forceinline__ float wave_sum32(float x) {
    v2f a; a[0] = x;    a[1] = 0.0f;
    v2f b; b[0] = 1.0f; b[1] = 1.0f;
    v8f c = {};
    c = __builtin_amdgcn_wmma_f32_16x16x4_f32(false, a, false, b,
                                              (short)0, c, false, false);
    float t = c[0] + c[1] + c[2] + c[3] + c[4] + c[5] + c[6] + c[7];
    t += __shfl_xor(t, 16, 32);
    return t;   // every lane holds the full wave sum
}

// stats layout (u32 slots at d_ws + npix*4):
// [0] f_nvalid  [1] f_nsel  [2] f_sum_above  [3] f_sum_gt  [4] f_cnt_gt
// [5] u_prefix  [6] u_krem  [7] pad          [8 .. 8+4*256) hist0..hist3
#define STATS_WORDS (8 + 4 * 256)

__global__ void zero_stats(unsigned* stats_u) {
    int i = blockIdx.x * blockDim.x + threadIdx.x;
    if (i < STATS_WORDS) stats_u[i] = 0u;
}

// Pass 1: CE per pixel, write CE scratch, accumulate branch-A stats,
// and build histogram of the top 8 bits (radix pass 0) in the same sweep.
__global__ void ce_kernel(const float* __restrict__ logits,
                          const int*   __restrict__ labels,
                          float*       __restrict__ ce,
                          float*       __restrict__ stats_f,
                          unsigned*    __restrict__ stats_u,
                          int npix, int chanStride) {
    __shared__ unsigned lhist[256];
    lhist[threadIdx.x] = 0u;
    __syncthreads();

    float cnt_valid = 0.f, cnt_sel = 0.f, sum_above = 0.f;
    const int stride = gridDim.x * blockDim.x;
    for (int p = blockIdx.x * blockDim.x + threadIdx.x; p < npix; p += stride) {
        const int b  = p / chanStride;
        const int hw = p - b * chanStride;
        const float* base = logits + (size_t)b * NCLASS * chanStride + hw;
        float v[NCLASS];
        float m = -3.4e38f;
        #pragma unroll
        for (int c = 0; c < NCLASS; ++c) {
            v[c] = base[(size_t)c * chanStride];
            m = fmaxf(m, v[c]);
        }
        float s = 0.f;
        #pragma unroll
        for (int c = 0; c < NCLASS; ++c) s += __expf(v[c] - m);
        const int  lab   = labels[p];
        const bool valid = (lab != IGNORE_L);
        const int  sl    = valid ? lab : 0;
        float cev = __logf(s) + (m - v[sl]);   // lse - logit[label] >= 0
        cev = valid ? cev : 0.0f;
        ce[p] = cev;
        if (valid) {
            cnt_valid += 1.f;
            if (cev > THRESH_F) { cnt_sel += 1.f; sum_above += cev; }
        }
        atomicAdd(&lhist[__float_as_uint(cev) >> 24], 1u);
    }
    __syncthreads();
    {   // merge radix pass-0 histogram
        unsigned cgt = lhist[threadIdx.x];
        if (cgt) atomicAdd(&stats_u[8 + threadIdx.x], cgt);
    }
    // wave reductions on the matrix pipe (EXEC all ones here)
    float tv = wave_sum32(cnt_valid);
    float ts = wave_sum32(cnt_sel);
    float ta = wave_sum32(sum_above);
    if ((threadIdx.x & 31) == 0) {
        atomicAdd(&stats_f[0], tv);
        atomicAdd(&stats_f[1], ts);
        atomicAdd(&stats_f[2], ta);
    }
}

// Radix histogram pass p (1..3): histogram next 8 bits of CE values whose
// top 8p bits match the current prefix.
__global__ void hist_kernel(const float* __restrict__ ce,
                            unsigned*    __restrict__ stats_u,
                            int npix, int pass) {
    __shared__ unsigned lhist[256];
    lhist[threadIdx.x] = 0u;
    __syncthreads();
    const unsigned pref      = stats_u[5];
    const int      shiftPref = 32 - 8 * pass;
    const int      shiftDig  = 24 - 8 * pass;
    const int      stride    = gridDim.x * blockDim.x;
    for (int p = blockIdx.x * blockDim.x + threadIdx.x; p < npix; p += stride) {
        const unsigned u = __float_as_uint(ce[p]);
        if ((u >> shiftPref) == pref)
            atomicAdd(&lhist[(u >> shiftDig) & 0xFFu], 1u);
    }
    __syncthreads();
    unsigned cgt = lhist[threadIdx.x];
    if (cgt) atomicAdd(&stats_u[8 + pass * 256 + threadIdx.x], cgt);
}

// Scan one 256-bin histogram from the top to refine (prefix, k_rem).
__global__ void select_kernel(unsigned* stats_u, int pass) {
    if (threadIdx.x != 0 || blockIdx.x != 0) return;
    const float* stats_f = (const float*)stats_u;
    unsigned k, pref;
    if (pass == 0) {
        const unsigned nvalid = (unsigned)(stats_f[0] + 0.5f);
        k = nvalid / 16u;                 // n_min (1-based rank of pivot)
        if (k == 0u) k = 1u;              // degenerate guard
        pref = 0u;
    } else {
        k    = stats_u[6];
        pref = stats_u[5];
    }
    const unsigned* h = stats_u + 8 + pass * 256;
    int d = 255;
    for (; d > 0; --d) {
        const unsigned c = h[d];
        if (k <= c) break;
        k -= c;
    }
    stats_u[5] = (pref << 8) | (unsigned)d;
    stats_u[6] = k;
}

// Sum / count of CE strictly greater than the exact pivot bit pattern.
__global__ void sumgt_kernel(const float*    __restrict__ ce,
                             float*          __restrict__ stats_f,
                             const unsigned* __restrict__ stats_u,
                             int npix) {
    const unsigned pivot = stats_u[5];
    float s = 0.f, cnt = 0.f;
    const int stride = gridDim.x * blockDim.x;
    for (int p = blockIdx.x * blockDim.x + threadIdx.x; p < npix; p += stride) {
        const float x = ce[p];
        if (__float_as_uint(x) > pivot) { s += x; cnt += 1.f; }
    }
    float ts = wave_sum32(s);
    float tc = wave_sum32(cnt);
    if ((threadIdx.x & 31) == 0) {
        atomicAdd(&stats_f[3], ts);
        atomicAdd(&stats_f[4], tc);
    }
}

__global__ void final_kernel(const unsigned* __restrict__ stats_u,
                             float* __restrict__ out) {
    if (threadIdx.x != 0 || blockIdx.x != 0) return;
    const float* sf = (const float*)stats_u;
    const unsigned nvalid = (unsigned)(sf[0] + 0.5f);
    const unsigned nmin   = nvalid / 16u;
    const unsigned nsel   = (unsigned)(sf[1] + 0.5f);
    const float mean_above = sf[2] / (float)(nsel > 1u ? nsel : 1u);
    const float pivot   = __uint_as_float(stats_u[5]);
    const float sumtopk = sf[3] + ((float)nmin - sf[4]) * pivot;
    const float mean_topk = sumtopk / (float)(nmin > 1u ? nmin : 1u);
    // cond = loss[n_min] > THRESH  <=>  n_sel > n_min  (sorted desc, exact w/ ties)
    out[0] = (nsel > nmin) ? mean_above : mean_topk;
}

extern "C" void kernel_launch(void* const* d_in, const int* in_sizes, int n_in,
                              void* d_out, int out_size, void* d_ws, size_t ws_size,
                              hipStream_t stream) {
    const float* logits = (const float*)d_in[0];
    const int*   labels = (const int*)d_in[1];
    const int npix       = in_sizes[1];        // 16*512*1024 = 8,388,608
    const int chanStride = 512 * 1024;         // H*W per reference layout

    float*    ce      = (float*)d_ws;
    unsigned* stats_u = (unsigned*)((char*)d_ws + (size_t)npix * sizeof(float));
    float*    stats_f = (float*)stats_u;

    const int TB = 256;
    const int NB = 8192;   // grid-stride, ~4 px/thread

    zero_stats<<<(STATS_WORDS + TB - 1) / TB, TB, 0, stream>>>(stats_u);
    ce_kernel<<<NB, TB, 0, stream>>>(logits, labels, ce, stats_f, stats_u,
                                     npix, chanStride);
    select_kernel<<<1, 32, 0, stream>>>(stats_u, 0);
    for (int pass = 1; pass <= 3; ++pass) {
        hist_kernel<<<NB, TB, 0, stream>>>(ce, stats_u, npix, pass);
        select_kernel<<<1, 32, 0, stream>>>(stats_u, pass);
    }
    sumgt_kernel<<<NB, TB, 0, stream>>>(ce, stats_f, stats_u, npix);
    final_kernel<<<1, 32, 0, stream>>>(stats_u, (float*)d_out);
}